// GrahamLoss_47425028882921
// MI455X (gfx1250) — compile-verified
//
#include <hip/hip_runtime.h>

// ---------------------------------------------------------------------------
// Gram-matrix style loss:  loss[b] = sum((Fd Fd^T - Fe Fe^T)^2) / denom
// B=16, C=128, HW=16384.  Memory-bound: 268MB @ 23.3TB/s ~ 11.5us floor.
// Strategy: single-pass fused diff-gram via V_WMMA_F32_16X16X4_F32 with a
// negated enc A-fragment; double-buffered LDS fed by GLOBAL_LOAD_ASYNC_TO_LDS
// (ASYNCcnt-tracked) so HBM streaming overlaps the WMMA pipeline.
// ---------------------------------------------------------------------------

typedef float v2f __attribute__((ext_vector_type(2)));
typedef float v4f __attribute__((ext_vector_type(4)));
typedef float v8f __attribute__((ext_vector_type(8)));
typedef int   v4i __attribute__((ext_vector_type(4)));

#define BATCH 16
#define CCH   128
#define HW    16384
#define KC    32                 // K-chunks per batch -> 512 workgroups total
#define KT    (HW / KC)          // 512 columns per workgroup
#define KSLAB 64                 // columns staged in LDS per iteration
#define LDS_STRIDE 68            // 64 + 4 pad: conflict-free, 16B-aligned rows
#define SLAB_F (CCH * LDS_STRIDE)

#define AS1 __attribute__((address_space(1)))
#define AS3 __attribute__((address_space(3)))

#if defined(__HIP_DEVICE_COMPILE__) && \
    __has_builtin(__builtin_amdgcn_global_load_async_to_lds_b128)
#define HAS_ASYNC_LDS 1
#else
#define HAS_ASYNC_LDS 0
#endif

__device__ __forceinline__ AS1 v4i* to_global(const void* p) {
  // integer round-trip: avoids addrspacecast legality questions and
  // drops constness (builtin takes a non-const AS1 pointer).
  return (AS1 v4i*)(unsigned long long)(size_t)p;
}
__device__ __forceinline__ AS3 v4i* to_lds_p(void* p) {
  // Flat addresses into the LDS aperture carry the LDS byte offset in the
  // low 32 bits, which is exactly what an AS(3) pointer holds.
  return (AS3 v4i*)(unsigned)(size_t)p;
}

template <int N>
__device__ __forceinline__ void wait_async_le() {
#if HAS_ASYNC_LDS
#if __has_builtin(__builtin_amdgcn_s_wait_asynccnt)
  __builtin_amdgcn_s_wait_asynccnt(N);
#else
  asm volatile("s_wait_asynccnt %0" ::"i"(N) : "memory");
#endif
#endif
}

// Stage one 128x64 slab of both matrices into LDS.
__device__ __forceinline__ void stage_slab(const float* __restrict__ enc,
                                           const float* __restrict__ dec,
                                           float* sEnc, float* sDec,
                                           long long matBase, int k0, int tid) {
#pragma unroll
  for (int r = 0; r < 8; ++r) {
    int slot = r * 256 + tid;              // 0..2047 float4 slots
    int row  = slot >> 4;                  // 16 float4 per row
    int colq = slot & 15;
    long long g = matBase + (long long)row * HW + k0 + colq * 4;
    int l = row * LDS_STRIDE + colq * 4;
#if HAS_ASYNC_LDS
    __builtin_amdgcn_global_load_async_to_lds_b128(
        to_global(enc + g), to_lds_p(sEnc + l), 0, 0);
    __builtin_amdgcn_global_load_async_to_lds_b128(
        to_global(dec + g), to_lds_p(sDec + l), 0, 0);
#else
    v4f ve = *(const v4f*)(enc + g);
    v4f vd = *(const v4f*)(dec + g);
    *(v4f*)(sEnc + l) = ve;
    *(v4f*)(sDec + l) = vd;
#endif
  }
}

__global__ void __launch_bounds__(256)
zero_ws_kernel(float* __restrict__ ws, int n) {
  int i = blockIdx.x * blockDim.x + threadIdx.x;
  if (i < n) ws[i] = 0.0f;
}

__global__ void __launch_bounds__(256)
gram_diff_kernel(const float* __restrict__ enc, const float* __restrict__ dec,
                 float* __restrict__ gdiff) {
  extern __shared__ float smem[];
  float* sE0 = smem;                 // buf0: enc slab
  float* sD0 = smem + SLAB_F;        // buf0: dec slab
  float* sE1 = smem + 2 * SLAB_F;    // buf1: enc slab
  float* sD1 = smem + 3 * SLAB_F;    // buf1: dec slab

  const int b    = blockIdx.x / KC;
  const int kc   = blockIdx.x % KC;
  const int tid  = threadIdx.x;
  const int lane = tid & 31;
  const int wave = tid >> 5;                 // 0..7: this wave's 16-row block i

  const long long matBase = (long long)b * CCH * HW;
  const int k_begin = kc * KT;

  // Per-lane fragment offset inside a 16-row block:
  //   lanes 0-15  -> row = lane,    k sub-offset 0 (holds K=k0,k0+1)
  //   lanes 16-31 -> row = lane-16, k sub-offset 2 (holds K=k0+2,k0+3)
  // (identical indexing for the f32 A-frag and for B = F^T frag)
  const int rb = (lane & 15) * LDS_STRIDE + ((lane >> 4) << 1);

  v8f acc[8] = {};   // 8 column tiles of the 16x16 diff-gram, f32 accum

  const int nSlabs = KT / KSLAB;             // 8

  stage_slab(enc, dec, sE0, sD0, matBase, k_begin, tid);

  for (int ks = 0; ks < nSlabs; ++ks) {
    float* cE = (ks & 1) ? sE1 : sE0;
    float* cD = (ks & 1) ? sD1 : sD0;
    float* nE = (ks & 1) ? sE0 : sE1;
    float* nD = (ks & 1) ? sD0 : sD1;

    if (ks + 1 < nSlabs) {
      // Kick off the next slab, then wait until only those 16 async ops
      // (per thread) remain in flight -> current slab fully landed in LDS.
      stage_slab(enc, dec, nE, nD, matBase, k_begin + (ks + 1) * KSLAB, tid);
      wait_async_le<16>();
    } else {
      wait_async_le<0>();
    }
    __syncthreads();

    // --- 16 K-steps of 4; 16 WMMAs per step per wave
    const int aBase = wave * 16 * LDS_STRIDE + rb;
#pragma unroll
    for (int kk = 0; kk < KSLAB; kk += 4) {
      v2f ad  =  *(const v2f*)(cD + aBase + kk);     // A frag (dec)
      v2f aeN = -*(const v2f*)(cE + aBase + kk);     // A frag (enc), negated
#pragma unroll
      for (int j = 0; j < 8; ++j) {
        const int bBase = j * 16 * LDS_STRIDE + rb + kk;
        v2f bd = *(const v2f*)(cD + bBase);          // B frag = Fd^T tile j
        v2f be = *(const v2f*)(cE + bBase);          // B frag = Fe^T tile j
        // acc += Ad*Bd ; acc += (-Ae)*Be   =>  acc accumulates Gd - Ge
        acc[j] = __builtin_amdgcn_wmma_f32_16x16x4_f32(
            false, ad,  false, bd, (short)0, acc[j], false, false);
        acc[j] = __builtin_amdgcn_wmma_f32_16x16x4_f32(
            false, aeN, false, be, (short)0, acc[j], false, false);
      }
    }
    __syncthreads();   // protects cur buffer before it is re-staged at ks+2
  }

  // --- reduce partial diff-gram tiles across K-chunks via f32 atomics
  // C/D layout: VGPR v, lanes 0-15 -> M=v, N=lane ; lanes 16-31 -> M=8+v
  const int m_off = (lane >> 4) * 8;
  const int n     = lane & 15;
  float* gb = gdiff + (long long)b * CCH * CCH;
#pragma unroll
  for (int j = 0; j < 8; ++j) {
#pragma unroll
    for (int v = 0; v < 8; ++v) {
      int row = wave * 16 + m_off + v;
      int col = j * 16 + n;
      atomicAdd(gb + row * CCH + col, acc[j][v]);
    }
  }
}

__global__ void __launch_bounds__(256)
reduce_loss_kernel(const float* __restrict__ gdiff, float* __restrict__ out) {
  const int b = blockIdx.x;
  const float* g = gdiff + (long long)b * CCH * CCH;
  float s = 0.0f;
  for (int i = threadIdx.x; i < CCH * CCH; i += 256) {
    float d = g[i];
    s += d * d;
  }
#pragma unroll
  for (int o = 16; o > 0; o >>= 1) s += __shfl_down(s, o, 32);
  __shared__ float wsum[8];
  if ((threadIdx.x & 31) == 0) wsum[threadIdx.x >> 5] = s;
  __syncthreads();
  if (threadIdx.x == 0) {
    float t = 0.0f;
#pragma unroll
    for (int w = 0; w < 8; ++w) t += wsum[w];
    // denom = 4 * (H*W)^2 * C^2
    const float inv_denom =
        1.0f / (4.0f * 16384.0f * 16384.0f * 128.0f * 128.0f);
    out[b] = t * inv_denom;
  }
}

extern "C" void kernel_launch(void* const* d_in, const int* in_sizes, int n_in,
                              void* d_out, int out_size, void* d_ws, size_t ws_size,
                              hipStream_t stream) {
  const float* feat  = (const float*)d_in[0];
  const float* featd = (const float*)d_in[1];
  float* gdiff = (float*)d_ws;                       // [B,128,128] f32 = 1 MB
  float* out   = (float*)d_out;                      // [B] f32

  const int nG = BATCH * CCH * CCH;
  zero_ws_kernel<<<(nG + 255) / 256, 256, 0, stream>>>(gdiff, nG);

  const size_t ldsBytes = 4u * SLAB_F * sizeof(float);   // 139,264 B (2 bufs)
  gram_diff_kernel<<<BATCH * KC, 256, ldsBytes, stream>>>(feat, featd, gdiff);

  reduce_loss_kernel<<<BATCH, 256, 0, stream>>>(gdiff, out);
}